// SwitchAggregator_12421045420199
// MI455X (gfx1250) — compile-verified
//
#include <hip/hip_runtime.h>
#include <hip/hip_bf16.h>

// Problem constants (match reference): E=16, B=8, S=2048, D=2048
#define NTOK    16384       // B*S
#define DIM     2048        // D
#define CAP     1024        // NTOK / E
#define NEXP    16
#define CHUNK   128         // tokens per chunk for the rank scan
#define NCHUNK  (NTOK / CHUNK)
#define HALFTOK 8192        // NTOK/2 : token t pairs with t+8192 in threefry counter space
#define HALFELT 16777216u   // NTOK*DIM/2

// keep <=> uniform(bits) < 0.9 <=> ((bits>>9)|0x3f800000) as f32 < (1.0f+0.9f)
// Both are positive normal floats, so fcmp == unsigned bit-pattern compare:
// threshold bits 0x3FF33333  =>  (bits>>9) < 0x733333  =>  bits < 0xE6666600.
#define KEEP_THRESH 0xE6666600u

typedef float __attribute__((ext_vector_type(4))) f32x4;   // native vector for NT builtins

// ---------------- threefry2x32 (exact JAX implementation, key = (0,42)) -------------
__device__ __forceinline__ unsigned tf_rotl(unsigned x, int r) {
    return (x << r) | (x >> (32 - r));
}

__device__ __forceinline__ void threefry2x32_0_42(unsigned c0, unsigned c1,
                                                  unsigned& o0, unsigned& o1) {
    const unsigned k0 = 0u, k1 = 42u;
    const unsigned ks2 = 0x1BD11BDAu ^ k0 ^ k1;
    unsigned x0 = c0 + k0;
    unsigned x1 = c1 + k1;
#define TF_G(r0, r1, r2, r3)                                    \
    x0 += x1; x1 = tf_rotl(x1, r0); x1 ^= x0;                   \
    x0 += x1; x1 = tf_rotl(x1, r1); x1 ^= x0;                   \
    x0 += x1; x1 = tf_rotl(x1, r2); x1 ^= x0;                   \
    x0 += x1; x1 = tf_rotl(x1, r3); x1 ^= x0;
    TF_G(13, 15, 26, 6)  x0 += k1;  x1 += ks2 + 1u;
    TF_G(17, 29, 16, 24) x0 += ks2; x1 += k0 + 2u;
    TF_G(13, 15, 26, 6)  x0 += k0;  x1 += k1 + 3u;
    TF_G(17, 29, 16, 24) x0 += k1;  x1 += ks2 + 4u;
    TF_G(13, 15, 26, 6)  x0 += ks2; x1 += k0 + 5u;
#undef TF_G
    o0 = x0; o1 = x1;
}

// ---------------- step 1: per-chunk expert histograms --------------------------------
__global__ void moe_hist_kernel(const int* __restrict__ routes, int* __restrict__ cnt) {
    __shared__ int h[NEXP];
    const int tid = threadIdx.x;                 // 128 threads
    if (tid < NEXP) h[tid] = 0;
    __syncthreads();
    atomicAdd(&h[routes[blockIdx.x * CHUNK + tid]], 1);
    __syncthreads();
    if (tid < NEXP) cnt[blockIdx.x * NEXP + tid] = h[tid];
}

// ---------------- step 2: per-expert exclusive scan over chunks ----------------------
__global__ void moe_scan_kernel(const int* __restrict__ cnt, int* __restrict__ base) {
    const int e = threadIdx.x;                   // one wave, 16 live lanes
    if (e < NEXP) {
        int run = 0;
        for (int c = 0; c < NCHUNK; ++c) {
            base[c * NEXP + e] = run;
            run += cnt[c * NEXP + e];
        }
    }
}

// ---------------- step 3: per-token within-expert rank -------------------------------
__global__ void moe_pos_kernel(const int* __restrict__ routes,
                               const int* __restrict__ base, int* __restrict__ pos) {
    __shared__ int r[CHUNK];
    const int tid = threadIdx.x;                 // 128 threads
    const int t = blockIdx.x * CHUNK + tid;
    r[tid] = routes[t];
    __syncthreads();
    const int my = r[tid];
    int rank = 0;
    for (int j = 0; j < tid; ++j) rank += (r[j] == my) ? 1 : 0;
    pos[t] = base[blockIdx.x * NEXP + my] + rank;
}

// ---------------- step 4: combine (gather + dropout + residual) ----------------------
// One block per token pair (t, t+8192); expert rows staged into LDS with CDNA5
// async global->LDS DMA (ASYNCcnt), all big streams marked non-temporal.
__global__ void __launch_bounds__(256)
moe_combine_kernel(const float* __restrict__ hidden, const float* __restrict__ expert,
                   const int* __restrict__ routes, const float* __restrict__ prob,
                   const int* __restrict__ pos, float* __restrict__ out) {
    __shared__ __align__(16) float smem[2 * DIM];   // rowA | rowB (16 KB)
    const int tid = threadIdx.x;                    // 256 threads
    const int tA = blockIdx.x;
    const int tB = tA + HALFTOK;

    const int rA = routes[tA], rB = routes[tB];
    const int pA = pos[tA],    pB = pos[tB];
    const float sA = prob[tA] / 0.9f;
    const float sB = prob[tB] / 0.9f;

    const float* rowA = expert + ((size_t)rA * CAP + (size_t)pA) * DIM;
    const float* rowB = expert + ((size_t)rB * CAP + (size_t)pB) * DIM;

#if defined(__gfx1250__)
    // CDNA5 async global->LDS DMA: 4 x b128 per thread (two 8KB rows per block).
    // Single-touch stream -> non-temporal hint.
    {
        const unsigned ldsA = (unsigned)(size_t)(void*)&smem[0];
        const unsigned ldsB = (unsigned)(size_t)(void*)&smem[DIM];
#pragma unroll
        for (int p = 0; p < 2; ++p) {
            const int k = p * 256 + tid;            // float4 index within row [0,512)
            unsigned long long gA = (unsigned long long)(const void*)(rowA + k * 4);
            unsigned long long gB = (unsigned long long)(const void*)(rowB + k * 4);
            unsigned lA = ldsA + (unsigned)(k * 16);
            unsigned lB = ldsB + (unsigned)(k * 16);
            asm volatile("global_load_async_to_lds_b128 %0, %1, off th:TH_LOAD_NT"
                         :: "v"(lA), "v"(gA) : "memory");
            asm volatile("global_load_async_to_lds_b128 %0, %1, off th:TH_LOAD_NT"
                         :: "v"(lB), "v"(gB) : "memory");
        }
    }
#if __has_builtin(__builtin_amdgcn_s_wait_asynccnt)
    __builtin_amdgcn_s_wait_asynccnt(0);
#else
    asm volatile("s_wait_asynccnt 0" ::: "memory");
#endif
#else
    // Non-gfx1250 fallback (host pass / other arch): plain staging copy.
#pragma unroll
    for (int p = 0; p < 2; ++p) {
        const int k = p * 256 + tid;
        ((f32x4*)smem)[k]         = ((const f32x4*)rowA)[k];
        ((f32x4*)(smem + DIM))[k] = ((const f32x4*)rowB)[k];
    }
#endif
    __syncthreads();

    const f32x4* hid4 = (const f32x4*)hidden;
    f32x4*       out4 = (f32x4*)out;
    const f32x4* smA  = (const f32x4*)&smem[0];
    const f32x4* smB  = (const f32x4*)&smem[DIM];

#pragma unroll
    for (int p = 0; p < 2; ++p) {
        const int j  = p * 256 + tid;               // float4 index within row [0,512)
        const int d0 = j * 4;
        const f32x4 a  = smA[j];
        const f32x4 b  = smB[j];
        const f32x4 hA = __builtin_nontemporal_load(&hid4[tA * (DIM / 4) + j]);
        const f32x4 hB = __builtin_nontemporal_load(&hid4[tB * (DIM / 4) + j]);

        f32x4 oa, ob;
#pragma unroll
        for (int e = 0; e < 4; ++e) {
            // flat mask index of (tA, d0+e); its threefry partner is (tB, d0+e)
            const unsigned c0 = (unsigned)(tA * DIM + d0 + e);
            unsigned o0, o1;
            threefry2x32_0_42(c0, c0 + HALFELT, o0, o1);
            const float kA = (o0 < KEEP_THRESH) ? sA : 0.0f;   // cmp_u32 + cndmask
            const float kB = (o1 < KEEP_THRESH) ? sB : 0.0f;
            oa[e] = fmaf(a[e], kA, hA[e]);
            ob[e] = fmaf(b[e], kB, hB[e]);
        }
        __builtin_nontemporal_store(oa, &out4[tA * (DIM / 4) + j]);
        __builtin_nontemporal_store(ob, &out4[tB * (DIM / 4) + j]);
    }
}

// ---------------- launcher -----------------------------------------------------------
extern "C" void kernel_launch(void* const* d_in, const int* in_sizes, int n_in,
                              void* d_out, int out_size, void* d_ws, size_t ws_size,
                              hipStream_t stream) {
    const float* hidden = (const float*)d_in[0];   // [B,S,D] f32
    const float* expert = (const float*)d_in[1];   // [E,C,D] f32
    const int*   routes = (const int*)d_in[2];     // [N] i32
    const float* prob   = (const float*)d_in[3];   // [N] f32
    float*       out    = (float*)d_out;           // [B,S,D] f32

    int* ws_pos  = (int*)d_ws;                     // NTOK ints
    int* ws_cnt  = ws_pos + NTOK;                  // NCHUNK*NEXP ints
    int* ws_base = ws_cnt + NCHUNK * NEXP;         // NCHUNK*NEXP ints

    moe_hist_kernel<<<NCHUNK, CHUNK, 0, stream>>>(routes, ws_cnt);
    moe_scan_kernel<<<1, 32, 0, stream>>>(ws_cnt, ws_base);
    moe_pos_kernel<<<NCHUNK, CHUNK, 0, stream>>>(routes, ws_base, ws_pos);
    moe_combine_kernel<<<HALFTOK, 256, 0, stream>>>(hidden, expert, routes, prob,
                                                    ws_pos, out);
}